// AggrSum_13288628814370
// MI455X (gfx1250) — compile-verified
//
#include <hip/hip_runtime.h>
#include <stdint.h>

// Segment-sum: out[v, :] += H[n, :] for all n with X_node[n] == v.
//   H:      [N, 64] float32   (512 MB, single-use stream -> non-temporal loads)
//   X_node: [N]     int32     (8 MB, single-use stream -> non-temporal load)
//   out:    [V, 64] float32   (25.6 MB accumulator -> keep L2-resident)
//
// Pure bandwidth problem: 520 MB / 23.3 TB/s ~= 22 us. No matmul structure,
// so no WMMA/TDM; the CDNA5-relevant choices are wave32 coalescing, hardware
// global_atomic_add_f32 (inline asm guarantees no CAS fallback), and cache
// temporal hints that reserve the 192 MB L2 for the atomic target.

#define D 64

typedef float f4 __attribute__((ext_vector_type(4)));

__global__ __launch_bounds__(256) void zero_f4_kernel(f4* __restrict__ out, int n4) {
    int i = blockIdx.x * 256 + threadIdx.x;
    if (i < n4) {
        // Regular-temporal store: leave zeroed accumulator lines dirty in L2
        // so the first wave of atomics hits in-cache.
        out[i] = (f4){0.f, 0.f, 0.f, 0.f};
    }
}

__global__ __launch_bounds__(256) void aggr_sum_scatter_kernel(
    const float* __restrict__ H,
    const int*   __restrict__ X_node,
    float*       __restrict__ out,
    int N)
{
    int tid = blockIdx.x * 256 + threadIdx.x;
    int row = tid >> 4;            // 16 lanes cooperate on one row of 64 floats
    if (row >= N) return;
    int col = (tid & 15) << 2;     // float4 slot within the row

    // Non-temporal 128-bit stream load: H is read exactly once; don't let the
    // 512 MB stream evict the L2-resident accumulator.
    const f4 h = __builtin_nontemporal_load(
        reinterpret_cast<const f4*>(H + (size_t)row * D + col));

    // Index is also a single-use stream.
    int v = __builtin_nontemporal_load(X_node + row);

    uint64_t addr = (uint64_t)(out + (size_t)v * D + col);

    // Four hardware f32 atomic adds to consecutive addresses (256B contiguous
    // span per 16-lane row group -> coalesces at L2). No-return form tracks
    // via STOREcnt; s_endpgm's implicit wait-idle covers completion.
    asm volatile(
        "global_atomic_add_f32 %0, %1, off\n\t"
        "global_atomic_add_f32 %0, %2, off offset:4\n\t"
        "global_atomic_add_f32 %0, %3, off offset:8\n\t"
        "global_atomic_add_f32 %0, %4, off offset:12"
        :
        : "v"(addr), "v"(h.x), "v"(h.y), "v"(h.z), "v"(h.w)
        : "memory");
}

extern "C" void kernel_launch(void* const* d_in, const int* in_sizes, int n_in,
                              void* d_out, int out_size, void* d_ws, size_t ws_size,
                              hipStream_t stream) {
    const float* H      = (const float*)d_in[0];
    const int*   X_node = (const int*)d_in[1];
    float*       out    = (float*)d_out;

    const int N = in_sizes[1];          // number of rows / messages

    // 1) Zero the output accumulator (out_size = V*D, divisible by 4).
    int n4 = out_size >> 2;
    int zblocks = (n4 + 255) / 256;
    zero_f4_kernel<<<zblocks, 256, 0, stream>>>((f4*)out, n4);

    // 2) Scatter-add: N rows * 16 lanes per row.
    long long total = (long long)N * 16;
    int sblocks = (int)((total + 255) / 256);
    aggr_sum_scatter_kernel<<<sblocks, 256, 0, stream>>>(H, X_node, out, N);
}